// CodebookDecoder_3040836846061
// MI455X (gfx1250) — compile-verified
//
#include <hip/hip_runtime.h>
#include <stdint.h>

#define B_   2
#define T_   4096
#define NB_  4
#define D_   512
#define E_   4096
#define NTOT (134217728u)   // B*T*NB*E
#define HALF_ (67108864u)   // NTOT/2

typedef __attribute__((ext_vector_type(2))) float v2f;
typedef __attribute__((ext_vector_type(8))) float v8f;
typedef unsigned int v4u __attribute__((ext_vector_type(4)));
typedef int v8i __attribute__((ext_vector_type(8)));
typedef int v4i __attribute__((ext_vector_type(4)));

// GEMM tiling: block = 256 threads = 8 waves (2 wave rows x 4 wave cols).
// Block tile: 64t x 256e. Wave tile: 32t x 64e -> 8 accumulators; each B
// fragment feeds 2 WMMAs (B global traffic halved vs 16t tiles).
// A tile (64t x 128d chunk) staged in LDS by the Tensor Data Mover with LDS
// padding: pad 4 DWORDs every 128 DWORDs -> row pitch 132 DWORDs; banks for
// the 16 rows x 4 K-offsets a fragment touches are (4r + j) mod 64: bijective,
// so the ds_load_b64 fragment reads are bank-conflict-free.
#define KCHUNK   128
#define A_PITCH  132                 // 128 data DWORDs + 4 pad DWORDs per row
#define A_ROWS   64

// ---------------------------------------------------------------------------
// JAX threefry2x32 noise: noise(i) = 1 - ( bitcast((bits>>9)|0x3f800000) - 1 )
// key = jax.random.key(42) -> (k0,k1) = (0,42); counter split: lower half is
// x0 of threefry(i, i+N/2), upper half is x1 of threefry(i-N/2, i).
// ---------------------------------------------------------------------------
__device__ __forceinline__ uint32_t rotl32(uint32_t x, uint32_t r) {
    return (x << r) | (x >> (32u - r));
}

__device__ __forceinline__ float noise_at(uint32_t i) {
    uint32_t c0, c1; int hi;
    if (i < HALF_) { c0 = i;         c1 = i + HALF_; hi = 0; }
    else           { c0 = i - HALF_; c1 = i;         hi = 1; }
    const uint32_t k0 = 0u, k1 = 42u, k2 = 0u ^ 42u ^ 0x1BD11BDAu;
    uint32_t x0 = c0 + k0, x1 = c1 + k1;
#define TFR(r) { x0 += x1; x1 = rotl32(x1, (r)); x1 ^= x0; }
    TFR(13) TFR(15) TFR(26) TFR(6)   x0 += k1; x1 += k2 + 1u;
    TFR(17) TFR(29) TFR(16) TFR(24)  x0 += k2; x1 += k0 + 2u;
    TFR(13) TFR(15) TFR(26) TFR(6)   x0 += k0; x1 += k1 + 3u;
    TFR(17) TFR(29) TFR(16) TFR(24)  x0 += k1; x1 += k2 + 4u;
    TFR(13) TFR(15) TFR(26) TFR(6)   x0 += k2; x1 += k0 + 5u;
#undef TFR
    uint32_t bits = hi ? x1 : x0;
    float u = __uint_as_float((bits >> 9) | 0x3f800000u) - 1.0f;
    return 1.0f - u;   // TEMP = 1.0
}

// ---------------------------------------------------------------------------
// TDM issue: 2D tile (KCHUNK cols x A_ROWS rows of f32), row stride NB*D
// elements in global memory, padded row-major into LDS. D# per ISA 08_async.
// ---------------------------------------------------------------------------
__device__ __forceinline__ void tdm_load_a_chunk(uint32_t lds_off, uint64_t gaddr) {
    v4u g0;
    g0[0] = 1u;                                   // count=1, user descriptor
    g0[1] = lds_off;                              // lds_addr (bytes)
    g0[2] = (uint32_t)gaddr;                      // global_addr[31:0]
    g0[3] = ((uint32_t)(gaddr >> 32) & 0x01FFFFFFu) | 0x80000000u; // addr[56:32] | type=2

    v8i g1;
    g1[0] = (int)((2u << 16)        // data_size = 4B
                | (1u << 20)        // pad_enable
                | (6u << 22)        // pad_interval: 128 DWORDs
                | (3u << 25));      // pad_amount:   4 DWORDs
    g1[1] = (int)((uint32_t)KCHUNK << 16);   // tensor_dim0[15:0]
    g1[2] = (int)((uint32_t)A_ROWS << 16);   // tensor_dim0[31:16]=0, tensor_dim1[15:0]
    g1[3] = (int)((uint32_t)KCHUNK << 16);   // tensor_dim1[31:16]=0, tile_dim0
    g1[4] = A_ROWS;                          // tile_dim1, tile_dim2=0
    g1[5] = NB_ * D_;                        // tensor_dim0_stride[31:0] = 2048 elements
    g1[6] = 0;                               // stride hi, tensor_dim1_stride lo
    g1[7] = 0;

    v4i gz = {0, 0, 0, 0};
#if __has_include(<hip/amd_detail/amd_gfx1250_TDM.h>)
    v8i gz8 = {0, 0, 0, 0, 0, 0, 0, 0};
    __builtin_amdgcn_tensor_load_to_lds(g0, g1, gz, gz, gz8, 0);
#else
    __builtin_amdgcn_tensor_load_to_lds(g0, g1, gz, gz, 0);
#endif
}

// ---------------------------------------------------------------------------
// Kernel 1: dist_logits via V_WMMA_F32_16X16X4_F32, A staged by TDM in LDS.
// A (16x4 f32): lanes 0-15 hold M=lane, K={k,k+1}; lanes 16-31 hold K={k+2,k+3}.
// C/D: VGPR r, lanes 0-15 -> (M=r, N=lane), lanes 16-31 -> (M=r+8, N=lane-16).
// ---------------------------------------------------------------------------
__global__ __launch_bounds__(256) void gemm_logits_kernel(
    const float* __restrict__ X,     // [B, T, NB*D]
    const float* __restrict__ W,     // [NB, E, D]
    float* __restrict__ outLog)      // [B, T, NB, E]
{
    __shared__ float sA[A_ROWS * A_PITCH];   // 33792 bytes

    const int lane   = threadIdx.x & 31;
    const int wave   = threadIdx.x >> 5;
    const int wave_t = wave >> 2;            // 0..1  -> 32 rows each
    const int wave_e = wave & 3;             // 0..3  -> 64 cols each
    const int bn     = blockIdx.z;
    const int b      = bn >> 2;
    const int n      = bn & 3;
    const int t_blk  = blockIdx.y << 6;                        // 64 rows / block
    const int e_base = ((int)blockIdx.x << 8) + (wave_e << 6); // 64 cols / wave

    const int koff = (lane >> 4) << 1;       // 0 for lanes 0-15, 2 for lanes 16-31
    const int tl   = (wave_t << 5) + (lane & 15);              // LDS row of subtile 0

    // B row pointers for the 4 e-subtiles of this wave
    const float* brow0 = W + ((size_t)n * E_ + e_base +  0 + (lane & 15)) * D_ + koff;
    const float* brow1 = W + ((size_t)n * E_ + e_base + 16 + (lane & 15)) * D_ + koff;
    const float* brow2 = W + ((size_t)n * E_ + e_base + 32 + (lane & 15)) * D_ + koff;
    const float* brow3 = W + ((size_t)n * E_ + e_base + 48 + (lane & 15)) * D_ + koff;

    const float* aChunkBase = X + (size_t)(b * T_ + t_blk) * (NB_ * D_) + n * D_;
    const uint32_t lds_off  = (uint32_t)(uintptr_t)&sA[0];

    v8f acc[2][4];
#pragma unroll
    for (int g = 0; g < 2; ++g)
#pragma unroll
        for (int j = 0; j < 4; ++j)
            acc[g][j] = (v8f){};

    for (int kc = 0; kc < D_; kc += KCHUNK) {
        if (wave == 0) {
            tdm_load_a_chunk(lds_off, (uint64_t)(uintptr_t)(aChunkBase + kc));
            __builtin_amdgcn_s_wait_tensorcnt(0);
        }
        __syncthreads();

#pragma unroll 4
        for (int k = 0; k < KCHUNK; k += 4) {
            v2f a0 = *(const v2f*)&sA[ tl       * A_PITCH + k + koff];  // ds_load_b64
            v2f a1 = *(const v2f*)&sA[(tl + 16) * A_PITCH + k + koff];
            v2f b0 = *(const v2f*)(brow0 + kc + k);
            v2f b1 = *(const v2f*)(brow1 + kc + k);
            v2f b2 = *(const v2f*)(brow2 + kc + k);
            v2f b3 = *(const v2f*)(brow3 + kc + k);
            acc[0][0] = __builtin_amdgcn_wmma_f32_16x16x4_f32(false, a0, false, b0, (short)0, acc[0][0], false, false);
            acc[0][1] = __builtin_amdgcn_wmma_f32_16x16x4_f32(false, a0, false, b1, (short)0, acc[0][1], false, false);
            acc[0][2] = __builtin_amdgcn_wmma_f32_16x16x4_f32(false, a0, false, b2, (short)0, acc[0][2], false, false);
            acc[0][3] = __builtin_amdgcn_wmma_f32_16x16x4_f32(false, a0, false, b3, (short)0, acc[0][3], false, false);
            acc[1][0] = __builtin_amdgcn_wmma_f32_16x16x4_f32(false, a1, false, b0, (short)0, acc[1][0], false, false);
            acc[1][1] = __builtin_amdgcn_wmma_f32_16x16x4_f32(false, a1, false, b1, (short)0, acc[1][1], false, false);
            acc[1][2] = __builtin_amdgcn_wmma_f32_16x16x4_f32(false, a1, false, b2, (short)0, acc[1][2], false, false);
            acc[1][3] = __builtin_amdgcn_wmma_f32_16x16x4_f32(false, a1, false, b3, (short)0, acc[1][3], false, false);
        }
        __syncthreads();
    }

    const int nc = e_base + (lane & 15);
#pragma unroll
    for (int g = 0; g < 2; ++g) {
        const int m0 = t_blk + (wave_t << 5) + (g << 4) + ((lane >> 4) << 3);
#pragma unroll
        for (int r = 0; r < 8; ++r) {
            const size_t rowbase = (size_t)((b * T_ + m0 + r) * NB_ + n) * E_;
            outLog[rowbase + nc +  0] = acc[g][0][r];
            outLog[rowbase + nc + 16] = acc[g][1][r];
            outLog[rowbase + nc + 32] = acc[g][2][r];
            outLog[rowbase + nc + 48] = acc[g][3][r];
        }
    }
}

// ---------------------------------------------------------------------------
// Kernel 2: per-column (b,n,e) top-3 over t (capacity = B*T/E*1.5 = 3).
// One thread per column; consecutive threads -> consecutive e (coalesced).
// Strict '>' insertion keeps the earliest t on ties (matches lax.top_k).
// ---------------------------------------------------------------------------
__global__ __launch_bounds__(256) void topk_cols_kernel(
    const float* __restrict__ logits,
    int* __restrict__ I0, int* __restrict__ I1, int* __restrict__ I2,
    float* __restrict__ V0, float* __restrict__ V1, float* __restrict__ V2)
{
    const int col = blockIdx.x * 256 + threadIdx.x;   // (b*NB + n)*E + e
    const int e   = col & (E_ - 1);
    const int n   = (col >> 12) & (NB_ - 1);
    const int b   = col >> 14;

    float v0 = -__builtin_inff(), v1 = v0, v2 = v0;
    int   i0 = 0, i1 = 0, i2 = 0;

    uint32_t idx = (uint32_t)(((b * T_) * NB_ + n) * E_ + e);
    const uint32_t stride = NB_ * E_;
    for (int t = 0; t < T_; ++t, idx += stride) {
        float v = logits[idx] * noise_at(idx);
        if (v > v0)      { v2 = v1; i2 = i1; v1 = v0; i1 = i0; v0 = v; i0 = t; }
        else if (v > v1) { v2 = v1; i2 = i1; v1 = v;  i1 = t; }
        else if (v > v2) { v2 = v;  i2 = t; }
    }
    I0[col] = i0; I1[col] = i1; I2[col] = i2;
    V0[col] = v0; V1[col] = v1; V2[col] = v2;
}

// ---------------------------------------------------------------------------
// Kernel 3: decoded_indices[b,t,n] = argmax_e (mask * noisy), first-occurrence.
// One wave32 per (b,t,n) row; masked value is top-k value when t is in the
// column's top-3 set, else 0 (the jnp zeros baseline). Shuffle reduction with
// value-desc / index-asc tie rule reproduces jnp.argmax semantics.
// ---------------------------------------------------------------------------
__global__ __launch_bounds__(256) void argmax_rows_kernel(
    const int* __restrict__ I0, const int* __restrict__ I1, const int* __restrict__ I2,
    const float* __restrict__ V0, const float* __restrict__ V1, const float* __restrict__ V2,
    int* __restrict__ outIdx)    // [B, T, NB]
{
    const int lane = threadIdx.x & 31;
    const int wave = threadIdx.x >> 5;
    const int row  = blockIdx.x * 8 + wave;    // (b*T + t)*NB + n
    const int n    = row & 3;
    const int t    = (row >> 2) & (T_ - 1);
    const int b    = row >> 14;
    const int colbase = (b * NB_ + n) * E_;

    float bestV = -__builtin_inff();
    int   bestE = 0x7fffffff;
    for (int j = 0; j < E_ / 32; ++j) {
        const int eidx = lane + (j << 5);
        const int c = colbase + eidx;
        float v = 0.0f;
        if      (I0[c] == t) v = V0[c];
        else if (I1[c] == t) v = V1[c];
        else if (I2[c] == t) v = V2[c];
        if (v > bestV) { bestV = v; bestE = eidx; }   // strict: first occurrence
    }
#pragma unroll
    for (int off = 16; off > 0; off >>= 1) {
        float ov = __shfl_xor(bestV, off, 32);
        int   oe = __shfl_xor(bestE, off, 32);
        if (ov > bestV || (ov == bestV && oe < bestE)) { bestV = ov; bestE = oe; }
    }
    if (lane == 0) outIdx[row] = bestE;
}

// ---------------------------------------------------------------------------
// Kernel 4: decoded_latents[b,t,n,:] = W[n, decoded_indices[b,t,n], :]
// float4 gather; W (32 MB) is L2-resident on MI455X (192 MB L2).
// ---------------------------------------------------------------------------
__global__ __launch_bounds__(256) void gather_latents_kernel(
    const float* __restrict__ W,
    const int* __restrict__ outIdx,
    float* __restrict__ outLat)      // [B, T, NB*D]
{
    const size_t gid  = (size_t)blockIdx.x * 256 + threadIdx.x;  // float4 units
    const size_t base = gid << 2;
    const int d  = (int)(base & (D_ - 1));
    const size_t rn = base >> 9;            // (b*T + t)*NB + n
    const int n  = (int)(rn & 3);
    const int idx = outIdx[rn];
    const float4 w = *(const float4*)(W + ((size_t)n * E_ + idx) * D_ + d);
    *(float4*)(outLat + base) = w;
}

// ---------------------------------------------------------------------------
// d_out layout (return order, flat): decoded_indices [32768] (int bits),
// decoded_latents [16777216] f32, dist_logits [134217728] f32.
// ---------------------------------------------------------------------------
extern "C" void kernel_launch(void* const* d_in, const int* /*in_sizes*/, int /*n_in*/,
                              void* d_out, int /*out_size*/, void* d_ws, size_t /*ws_size*/,
                              hipStream_t stream) {
    const float* X = (const float*)d_in[0];   // [B, T, NB*D]
    const float* W = (const float*)d_in[1];   // [NB, E, D]

    int*   outIdx = (int*)d_out;
    float* outLat = (float*)d_out + 32768;
    float* outLog = (float*)d_out + 32768 + 16777216;

    const int NCOL = B_ * NB_ * E_;           // 32768 columns
    int*   I0 = (int*)d_ws;
    int*   I1 = I0 + NCOL;
    int*   I2 = I1 + NCOL;
    float* V0 = (float*)(I2 + NCOL);
    float* V1 = V0 + NCOL;
    float* V2 = V1 + NCOL;

    // 1) fp32 WMMA GEMM with TDM-staged A tiles -> dist_logits (into d_out)
    dim3 g1(E_ / 256, T_ / 64, B_ * NB_);     // 8 waves/block, 32t x 64e per wave
    gemm_logits_kernel<<<g1, 256, 0, stream>>>(X, W, outLog);

    // 2) top-3 over T per (b,n,e) column, threefry noise recomputed on the fly
    topk_cols_kernel<<<NCOL / 256, 256, 0, stream>>>(outLog, I0, I1, I2, V0, V1, V2);

    // 3) argmax over E per (b,t,n) row -> decoded_indices
    argmax_rows_kernel<<<(B_ * T_ * NB_) / 8, 256, 0, stream>>>(I0, I1, I2, V0, V1, V2, outIdx);

    // 4) embedding gather -> decoded_latents
    gather_latents_kernel<<<(B_ * T_ * NB_ * D_ / 4) / 256, 256, 0, stream>>>(W, outIdx, outLat);
}